// iRadonModel_37005438223143
// MI455X (gfx1250) — compile-verified
//
#include <hip/hip_runtime.h>
#include <hip/hip_bf16.h>

#define DET     512
#define NA      180
#define OUT     362
#define RADIUS  181      // OUT/2
#define MT      12       // angle tiles: 180 padded to 192
#define NT      32       // detector tiles: 512/16
#define KSTEPS  128      // K=512 in steps of 4
#define PI_F    3.14159265358979323846f

typedef float v2f __attribute__((ext_vector_type(2)));
typedef float v8f __attribute__((ext_vector_type(8)));

// ---------------------------------------------------------------------------
// Kernel 1: build the spatial ramp-filter matrix H[k][n] = g(|n-k|).
// g(0)=0.5, g(d)=-2/(pi^2 d^2) for odd d, 0 for even d!=0.
// (Exactly equivalent to the reference's FFT * ramp * IFFT on the zero-padded
//  signal, since the frequency response is fft of a real, even kernel.)
// ---------------------------------------------------------------------------
__global__ void build_filter(float* __restrict__ H) {
    int idx = blockIdx.x * blockDim.x + threadIdx.x;
    if (idx >= DET * DET) return;
    int k = idx >> 9;
    int n = idx & (DET - 1);
    int d = n - k; if (d < 0) d = -d;
    float v = 0.0f;
    if (d == 0) {
        v = 0.5f;
    } else if (d & 1) {
        float fd = (float)d;
        v = (-2.0f / (PI_F * PI_F)) / (fd * fd);
    }
    H[idx] = v;
}

// ---------------------------------------------------------------------------
// Kernel 2: ft[b][a][m] = sum_k x[b][k][a] * H[k][m]   (H symmetric)
// One wave32 per 16x16 output tile, V_WMMA_F32_16X16X4_F32 over K=512.
// A fragment (16x4 f32, 2 VGPRs): lane&15 = M (angle row), K = 2*(lane>>4)+j.
// B fragment mirrored: lane&15 = N (detector col), K = 2*(lane>>4)+j.
// OOB angle rows (>=180) are zero-masked via multiply (no EXEC divergence,
// WMMA requires EXEC all-1s).
// ---------------------------------------------------------------------------
__global__ void __launch_bounds__(32)
wmma_filter_gemm(const float* __restrict__ x, const float* __restrict__ H,
                 float* __restrict__ ft) {
    int bid   = blockIdx.x;
    int b     = bid / (MT * NT);
    int r     = bid % (MT * NT);
    int mtile = r / NT;
    int ntile = r % NT;

    int lane  = threadIdx.x;         // 0..31
    int l16   = lane & 15;
    int khalf = (lane >> 4) << 1;    // 0 or 2

    int   arow  = mtile * 16 + l16;              // angle index for A rows
    int   arc   = arow < NA ? arow : (NA - 1);   // clamped (convergent load)
    float amask = arow < NA ? 1.0f : 0.0f;
    int   ncol  = ntile * 16 + l16;              // detector column for B/D

    const float* xb = x + (size_t)b * DET * NA;

    v8f acc = {0.f, 0.f, 0.f, 0.f, 0.f, 0.f, 0.f, 0.f};

#pragma unroll 4
    for (int kk = 0; kk < KSTEPS; ++kk) {
        int kb = (kk << 2) + khalf;
        v2f av = { xb[(kb    ) * NA + arc] * amask,
                   xb[(kb + 1) * NA + arc] * amask };
        v2f bv = { H[(kb    ) * DET + ncol],
                   H[(kb + 1) * DET + ncol] };
        // (neg_a, A, neg_b, B, c_mod, C, reuse_a, reuse_b)
        acc = __builtin_amdgcn_wmma_f32_16x16x4_f32(
                  false, av, false, bv, (short)0, acc, false, false);
    }

    // D layout: VGPR r holds M=r (lanes 0-15) / M=r+8 (lanes 16-31), N=lane&15
    int mbase = mtile * 16 + ((lane >> 4) << 3);
#pragma unroll
    for (int rr = 0; rr < 8; ++rr) {
        int a = mbase + rr;
        if (a < NA) ft[((size_t)b * NA + a) * DET + ncol] = acc[rr];
    }
}

// ---------------------------------------------------------------------------
// Kernel 3: backprojection. out[b,0,i,j] = pi/360 * sum_a interp(ft[b][a][.])
// t = ypr*cos - xpr*sin ; pos = t + 256 ; 2-tap linear interp, 0 outside.
// cos/sin table built once per block in LDS.
// ---------------------------------------------------------------------------
__global__ void backproj(const float* __restrict__ ft, float* __restrict__ out) {
    __shared__ float cs[NA];
    __shared__ float sn[NA];
    int t = threadIdx.x;
    if (t < NA) {
        float th = (float)t * (PI_F / 180.0f);
        cs[t] = cosf(th);
        sn[t] = sinf(th);
    }
    __syncthreads();

    long tid = (long)blockIdx.x * blockDim.x + threadIdx.x;
    const long total = 4L * OUT * OUT;
    if (tid >= total) return;

    int b   = (int)(tid / (OUT * OUT));
    int rem = (int)(tid % (OUT * OUT));
    float xp = (float)(rem / OUT - RADIUS);   // row coordinate (xpr)
    float yp = (float)(rem % OUT - RADIUS);   // col coordinate (ypr)

    const float* fb = ft + (size_t)b * NA * DET;
    float acc = 0.0f;

    for (int a = 0; a < NA; ++a) {
        float pos = yp * cs[a] - xp * sn[a] + (float)(DET / 2);
        float fl  = floorf(pos);
        int   i0  = (int)fl;
        float w   = pos - fl;
        int c0 = min(max(i0, 0), DET - 1);
        int c1 = min(max(i0 + 1, 0), DET - 1);
        const float* row = fb + a * DET;
        float g0 = row[c0];
        float g1 = row[c1];
        float v  = (1.0f - w) * g0 + w * g1;
        acc += (pos >= 0.0f && pos <= (float)(DET - 1)) ? v : 0.0f;
    }
    out[tid] = acc * (PI_F / 360.0f);
}

// ---------------------------------------------------------------------------
extern "C" void kernel_launch(void* const* d_in, const int* in_sizes, int n_in,
                              void* d_out, int out_size, void* d_ws, size_t ws_size,
                              hipStream_t stream) {
    const float* x = (const float*)d_in[0];       // (4,1,512,180) float32
    float* H  = (float*)d_ws;                     // 512*512 floats (1 MB)
    float* ft = H + (size_t)DET * DET;            // 4*180*512 floats (1.44 MB)
    float* out = (float*)d_out;                   // (4,1,362,362) float32

    // 1) filter matrix
    build_filter<<<(DET * DET + 255) / 256, 256, 0, stream>>>(H);

    // 2) ramp filtering as FP32 WMMA GEMM: 4 batches * 12 * 32 tiles, wave32
    wmma_filter_gemm<<<4 * MT * NT, 32, 0, stream>>>(x, H, ft);

    // 3) backprojection
    long total = 4L * OUT * OUT;
    backproj<<<(int)((total + 255) / 256), 256, 0, stream>>>(ft, out);
}